// RefWignerRotation_7232724927061
// MI455X (gfx1250) — compile-verified
//
#include <hip/hip_runtime.h>
#include <hip/hip_bf16.h>

typedef __attribute__((ext_vector_type(2))) float v2f;
typedef __attribute__((ext_vector_type(8))) float v8f;

#define NSAMP 8192
#define DIMR  592
#define NC    16

// Fixed J matrices (y<->z axis exchange in the real-SH basis), row-major.
// Verified against the reference generators: J*X1*J == X0 for l=1,2.
__constant__ float cJ[9 + 25 + 49] = {
  // J1 (3x3) at offset 0
  0.f, 1.f, 0.f,
  1.f, 0.f, 0.f,
  0.f, 0.f,-1.f,
  // J2 (5x5) at offset 9
  0.f, 0.f, 0.f,-1.f, 0.f,
  0.f, 1.f, 0.f, 0.f, 0.f,
  0.f, 0.f,-0.5f,0.f,-0.8660254037844386f,
 -1.f, 0.f, 0.f, 0.f, 0.f,
  0.f, 0.f,-0.8660254037844386f, 0.f, 0.5f,
  // J3 (7x7) at offset 34
  0.f, 0.f, 0.f,-0.7905694150420949f, 0.f, 0.6123724356957945f, 0.f,
  0.f,-1.f, 0.f, 0.f, 0.f, 0.f, 0.f,
  0.f, 0.f, 0.f,-0.6123724356957945f, 0.f,-0.7905694150420949f, 0.f,
 -0.7905694150420949f, 0.f,-0.6123724356957945f, 0.f, 0.f, 0.f, 0.f,
  0.f, 0.f, 0.f, 0.f, 0.25f, 0.f, 0.9682458365518543f,
  0.6123724356957945f, 0.f,-0.7905694150420949f, 0.f, 0.f, 0.f, 0.f,
  0.f, 0.f, 0.f, 0.f, 0.9682458365518543f, 0.f,-0.25f
};

// D^l(a,b,g) = Z(a) * (J * Z(b) * J) * Z(g), with
// Z(t)[i][i] = cos((L-i)t), Z(t)[i][2L-i] = sin((L-i)t).
template<int L>
__device__ inline void compute_D(float al, float be, float ga, float* __restrict__ out) {
  constexpr int d = 2 * L + 1;
  constexpr int joff = (L == 1) ? 0 : ((L == 2) ? 9 : 34);
  const float* J = &cJ[joff];

  float cb[d], sb[d], ca[d], sa[d], cg[d], sg[d];
#pragma unroll
  for (int p = 0; p < d; ++p) {
    const float f = (float)(L - p);
    sincosf(f * be, &sb[p], &cb[p]);
    sincosf(f * al, &sa[p], &ca[p]);
    sincosf(f * ga, &sg[p], &cg[p]);
  }

  // tmp = Z(b) * J
  float tmp[d][d];
#pragma unroll
  for (int p = 0; p < d; ++p)
#pragma unroll
    for (int j = 0; j < d; ++j)
      tmp[p][j] = cb[p] * J[p * d + j] + sb[p] * J[(d - 1 - p) * d + j];

  // W = J * tmp  (= expm(b*X0))
  float W[d][d];
#pragma unroll
  for (int i = 0; i < d; ++i)
#pragma unroll
    for (int j = 0; j < d; ++j) {
      float s = 0.f;
#pragma unroll
      for (int p = 0; p < d; ++p) s += J[i * d + p] * tmp[p][j];
      W[i][j] = s;
    }

  // D = Z(a) * W * Z(g), folded directly (no intermediate M matrix)
#pragma unroll
  for (int i = 0; i < d; ++i) {
    const int i2 = d - 1 - i;
#pragma unroll
    for (int j = 0; j < d; ++j) {
      const int j2 = d - 1 - j;
      const float m0 = ca[i] * W[i][j]  + sa[i] * W[i2][j];   // M[i][j]
      const float m1 = ca[i] * W[i][j2] + sa[i] * W[i2][j2];  // M[i][2L-j]
      out[i * d + j] = cg[j] * m0 - sg[j] * m1;
    }
  }
}

__global__ void wigner_compute_D(const float* __restrict__ alpha,
                                 const float* __restrict__ beta,
                                 const float* __restrict__ gamma,
                                 float* __restrict__ Dws) {
  const int n = blockIdx.x * blockDim.x + threadIdx.x;
  if (n >= NSAMP) return;
  const float al = alpha[n], be = beta[n], ga = gamma[n];
  float* o = Dws + (size_t)n * 84;   // D1:[0,9) D2:[9,34) D3:[34,83)
  compute_D<1>(al, be, ga, o);
  compute_D<2>(al, be, ga, o + 9);
  compute_D<3>(al, be, ga, o + 34);
}

// out[i,c] = sum_j D[i,j] * x[m*d+j, c] via V_WMMA_F32_16X16X4_F32.
// A = D (rows M=i, K=j; zero padded), loaded once per l, reused for all m.
// B = x rows (K=j, N=c): lane<16 -> row j0 (cols 0..15), lane>=16 -> row j0+2.
// C/D: VGPR v, lanes 0-15 hold row M=v -> contiguous 64B stores per row.
template<int L, int MUL, int OFF>
__device__ inline void apply_block(const float* __restrict__ xb, float* __restrict__ ob,
                                   const float* __restrict__ D, int lane) {
  constexpr int d  = 2 * L + 1;
  constexpr int nK = (d + 3) / 4;   // 1 for d=3, 2 for d=5,7
  const int i  = lane & 15;
  const int hi = lane >> 4;
  const int c  = i;

  v2f A[nK];
#pragma unroll
  for (int k = 0; k < nK; ++k) {
    const int j0 = k * 4 + hi * 2;
    A[k].x = (i < d && j0     < d) ? D[i * d + j0]     : 0.f;
    A[k].y = (i < d && j0 + 1 < d) ? D[i * d + j0 + 1] : 0.f;
  }

  for (int m = 0; m < MUL; ++m) {
    const int rb = OFF + m * d;
    v8f acc = {};
#pragma unroll
    for (int k = 0; k < nK; ++k) {
      const int j0 = k * 4 + hi * 2;
      int r0 = rb + j0;     if (r0 > DIMR - 1) r0 = DIMR - 1;  // padded j, weight=0
      int r1 = rb + j0 + 1; if (r1 > DIMR - 1) r1 = DIMR - 1;
      v2f B;
      B.x = xb[r0 * NC + c];
      B.y = xb[r1 * NC + c];
      // 8 args: (neg_a, A, neg_b, B, c_mod, C, reuse_a, reuse_b)
      acc = __builtin_amdgcn_wmma_f32_16x16x4_f32(
          false, A[k], false, B, (short)0, acc, false, false);
    }
    if (lane < 16) {
#pragma unroll
      for (int v = 0; v < d; ++v)
        ob[(rb + v) * NC + c] = acc[v];
    }
  }
}

__global__ void __launch_bounds__(256)
wigner_apply(const float* __restrict__ x, const float* __restrict__ Dws,
             float* __restrict__ out) {
  const int lane = threadIdx.x & 31;
  const int wave = threadIdx.x >> 5;
  const int n = blockIdx.x * 8 + wave;           // one wave per sample
  const float* xb = x   + (size_t)n * (DIMR * NC);
  float*       ob = out + (size_t)n * (DIMR * NC);
  const float* Dn = Dws + (size_t)n * 84;

  // l=0 (128 scalars x 16 ch): D = [1] -> pure streaming copy, b128 both ways.
  const float4* xs = (const float4*)xb;
  float4*       os = (float4*)ob;
#pragma unroll
  for (int t = 0; t < 16; ++t) os[t * 32 + lane] = xs[t * 32 + lane];

  apply_block<1, 64, 128>(xb, ob, Dn,      lane);  // 64 x 3-vectors
  apply_block<2, 32, 320>(xb, ob, Dn +  9, lane);  // 32 x 5-vectors
  apply_block<3, 16, 480>(xb, ob, Dn + 34, lane);  // 16 x 7-vectors
}

extern "C" void kernel_launch(void* const* d_in, const int* in_sizes, int n_in,
                              void* d_out, int out_size, void* d_ws, size_t ws_size,
                              hipStream_t stream) {
  const float* x     = (const float*)d_in[0];
  const float* alpha = (const float*)d_in[1];
  const float* beta  = (const float*)d_in[2];
  const float* gamma = (const float*)d_in[3];
  float* Dws = (float*)d_ws;  // 8192 * 84 * 4 B = 2.75 MB (fits ws / L2)

  wigner_compute_D<<<dim3((NSAMP + 255) / 256), dim3(256), 0, stream>>>(
      alpha, beta, gamma, Dws);
  wigner_apply<<<dim3(NSAMP / 8), dim3(256), 0, stream>>>(
      x, Dws, (float*)d_out);
}